// MultiScaleSSIMLoss_39264591020035
// MI455X (gfx1250) — compile-verified
//
#include <hip/hip_runtime.h>

// ---------------- MS-SSIM on gfx1250: fused per-scale kernel ----------------
// Separable 11-tap Gaussian blur with BOTH directions on the matrix pipes:
//   horizontal:  H(16x16)   = In(16x28)  x G(28x16)   (G in the B slot)
//   vertical  :  out(16x16) = G^T(16x28) x H(28x16)   (G in the A slot)
// where G is the banded Gaussian matrix g[k - c]; the per-lane register
// fragments for both slots are numerically identical, so one 7 x v2f set
// (14 VGPRs) is built once and reused for all 350 WMMAs per workgroup.

typedef __attribute__((ext_vector_type(2))) float v2f;
typedef __attribute__((ext_vector_type(8))) float v8f;

#define TW 32             // output tile width
#define THT 32            // output tile height
#define HALO 5
#define RGN (THT + 2*HALO)    // 42 = valid rows/cols of the haloed region
#define PRGN 48               // padded extent (3 x 16 row blocks), zero-filled
#define SIN_STRIDE 52     // A-reads: 52*L mod 64 distinct over the 16-row stripe
#define SH_STRIDE 44      // stores (+8 rows = 32 banks) and vertical B-reads
                          // (+2 rows = 24 banks) keep half-waves on disjoint banks
#define C1V 1.0e-4f       // 0.01^2
#define C2V 9.0e-4f       // 0.03^2

// Extended banded Gaussian: GEXT[16 + t] = g[t] for t in [0,10], zero elsewhere.
// g = normalized exp(-d^2 / (2*1.5^2)), d = -5..5.
__device__ __constant__ float GEXT[44] = {
  0.f,0.f,0.f,0.f, 0.f,0.f,0.f,0.f, 0.f,0.f,0.f,0.f, 0.f,0.f,0.f,0.f,      // 0..15
  0.00102836f, 0.00759870f, 0.03600077f, 0.10936089f, 0.21300551f,         // g0..g4
  0.26601172f,                                                             // g5
  0.21300551f, 0.10936089f, 0.03600077f, 0.00759870f, 0.00102836f,         // g6..g10
  0.f,0.f,0.f,0.f,0.f, 0.f,0.f,0.f,0.f,0.f, 0.f,0.f,0.f,0.f,0.f, 0.f,0.f  // 27..43
};

__launch_bounds__(128)
__global__ void ms_ssim_scale_kernel(const float* __restrict__ img1,
                                     const float* __restrict__ img2,
                                     float* __restrict__ acc,
                                     int H, int W)
{
  __shared__ float sIn1[PRGN][SIN_STRIDE];
  __shared__ float sIn2[PRGN][SIN_STRIDE];
  __shared__ float sH[5][PRGN][SH_STRIDE];   // rows 42..47 hold true zeros

  const int tid  = threadIdx.x;
  const int lane = tid & 31;
  const int wave = tid >> 5;            // 4 waves

  const int c0 = blockIdx.x * TW;
  const int r0 = blockIdx.y * THT;
  const size_t plane_off = (size_t)blockIdx.z * (size_t)H * (size_t)W;

  // Banded-Gaussian register fragments, shared by both passes.
  //   B slot (horizontal): B[k][c] = g[k - c]
  //   A slot (vertical)  : A[r][k] = g[k - r]
  // Per-lane value in both layouts: GEXT[4j + koff + {0,1} - (lane&15) + 16].
  const int low16 = lane & 15;
  const int koff  = (lane >> 4) << 1;   // 0 or 2
  v2f gfrag[7];
  #pragma unroll
  for (int j = 0; j < 7; ++j) {
    const int kb = 4 * j + koff;
    gfrag[j].x = GEXT[kb - low16 + 16];
    gfrag[j].y = GEXT[kb - low16 + 17];
  }

  // ---- Stage 1: zero-padded cooperative load over the full 48x48 extent ----
  {
    const int ty = tid >> 5;            // 0..3
    const int tx = tid & 31;
    for (int r = ty; r < PRGN; r += 4) {
      const int gr = r0 - HALO + r;
      const bool rok = (r < RGN) && (gr >= 0) && (gr < H);
      for (int c = tx; c < PRGN; c += 32) {
        const int gc = c0 - HALO + c;
        float v1 = 0.f, v2 = 0.f;
        if (rok && c < RGN && gc >= 0 && gc < W) {
          const size_t off = plane_off + (size_t)gr * W + gc;
          v1 = img1[off];
          v2 = img2[off];
        }
        sIn1[r][c] = v1;
        sIn2[r][c] = v2;
      }
    }
  }
  __syncthreads();

  // ---- Stage 2: horizontal blur of all 5 fields on WMMA ----
  // 6 subtiles (3 row-blocks x 2 col-blocks) over 4 waves; fields formed
  // elementwise on the A-fragments (x, y, x^2, y^2, x*y).
  for (int s = wave; s < 6; s += 4) {
    const int rb = (s >> 1) << 4;       // 0,16,32
    const int cb = (s & 1) << 4;        // 0,16
    const int arow = rb + low16;
    v8f h0 = {0.f,0.f,0.f,0.f,0.f,0.f,0.f,0.f};
    v8f h1 = h0, h2 = h0, h3 = h0, h4 = h0;
    #pragma unroll
    for (int j = 0; j < 7; ++j) {
      const int k = cb + 4 * j + koff;  // <= 43, zero-padded
      v2f x, y;
      x.x = sIn1[arow][k];
      x.y = sIn1[arow][k + 1];
      y.x = sIn2[arow][k];
      y.y = sIn2[arow][k + 1];
      const v2f xx = x * x;
      const v2f yy = y * y;
      const v2f xy = x * y;
      h0 = __builtin_amdgcn_wmma_f32_16x16x4_f32(false, x,  false, gfrag[j], (short)0, h0, false, false);
      h1 = __builtin_amdgcn_wmma_f32_16x16x4_f32(false, y,  false, gfrag[j], (short)0, h1, false, false);
      h2 = __builtin_amdgcn_wmma_f32_16x16x4_f32(false, xx, false, gfrag[j], (short)0, h2, false, false);
      h3 = __builtin_amdgcn_wmma_f32_16x16x4_f32(false, yy, false, gfrag[j], (short)0, h3, false, false);
      h4 = __builtin_amdgcn_wmma_f32_16x16x4_f32(false, xy, false, gfrag[j], (short)0, h4, false, false);
    }
    // D layout: VGPR i -> row i (lanes 0-15) / row i+8 (lanes 16-31)
    const int srow = rb + ((lane >> 4) << 3);
    const int scol = cb + low16;
    #pragma unroll
    for (int i = 0; i < 8; ++i) {
      sH[0][srow + i][scol] = h0[i];
      sH[1][srow + i][scol] = h1[i];
      sH[2][srow + i][scol] = h2[i];
      sH[3][srow + i][scol] = h3[i];
      sH[4][srow + i][scol] = h4[i];
    }
  }
  __syncthreads();

  // ---- Stage 3: vertical blur as banded matmul (G in the A slot) ----
  const int sx = wave & 1;              // 16x16 subtile coords in 32x32 tile
  const int sy = wave >> 1;
  const int colbase = sx * 16;
  const int rowbase = sy * 16;          // sH row corresponding to k = 0

  // B layout: lanes 0-15 rows (4j,4j+1), lanes 16-31 rows (4j+2,4j+3)
  const int bcol = colbase + low16;

  v8f blur[5];
  #pragma unroll
  for (int f = 0; f < 5; ++f) {
    v8f c8 = {0.f,0.f,0.f,0.f,0.f,0.f,0.f,0.f};
    #pragma unroll
    for (int j = 0; j < 7; ++j) {
      const int k0 = rowbase + 4 * j + koff;   // <= 43, rows 42..47 are zeros
      v2f b;
      b.x = sH[f][k0][bcol];
      b.y = sH[f][k0 + 1][bcol];
      c8 = __builtin_amdgcn_wmma_f32_16x16x4_f32(false, gfrag[j], false, b,
                                                 (short)0, c8, false, false);
    }
    blur[f] = c8;
  }

  // ---- SSIM map on the 8 accumulator elements, then wave reduce ----
  float psum = 0.f;
  #pragma unroll
  for (int i = 0; i < 8; ++i) {
    const float m1  = blur[0][i];
    const float m2  = blur[1][i];
    const float m1s = m1 * m1;
    const float m2s = m2 * m2;
    const float m12 = m1 * m2;
    const float s1  = blur[2][i] - m1s;
    const float s2  = blur[3][i] - m2s;
    const float s12 = blur[4][i] - m12;
    const float num = (2.f * m12 + C1V) * (2.f * s12 + C2V);
    const float den = (m1s + m2s + C1V) * (s1 + s2 + C2V);
    psum += num / den;
  }
  #pragma unroll
  for (int off = 16; off > 0; off >>= 1)
    psum += __shfl_xor(psum, off, 32);
  if (lane == 0)
    atomicAdd(acc, psum);
}

// ---- 2x2 average pool of both images (feeds next scale) ----
__launch_bounds__(256)
__global__ void down2_kernel(const float* __restrict__ in1,
                             const float* __restrict__ in2,
                             float* __restrict__ o1,
                             float* __restrict__ o2,
                             int H2, int W2)
{
  const int c = blockIdx.x * 64 + (threadIdx.x & 63);
  const int r = blockIdx.y * 4 + (threadIdx.x >> 6);
  const int p = blockIdx.z;
  const int W = W2 * 2;
  const size_t ib = ((size_t)p * (size_t)(H2 * 2) + (size_t)(2 * r)) * (size_t)W
                    + (size_t)(2 * c);
  const size_t ob = ((size_t)p * (size_t)H2 + (size_t)r) * (size_t)W2 + (size_t)c;
  o1[ob] = 0.25f * (in1[ib] + in1[ib + 1] + in1[ib + W] + in1[ib + W + 1]);
  o2[ob] = 0.25f * (in2[ib] + in2[ib + 1] + in2[ib + W] + in2[ib + W + 1]);
}

// ---- combine the 4 per-scale sums into the final loss ----
__global__ void finalize_kernel(const float* __restrict__ acc, float* __restrict__ out)
{
  if (threadIdx.x == 0 && blockIdx.x == 0) {
    const float wgt[4] = {0.0448f, 0.2856f, 0.3001f, 0.2363f};
    const float inv[4] = {1.f / 12582912.f, 1.f / 3145728.f,
                          1.f / 786432.f,  1.f / 196608.f};
    float loss = 0.f;
    #pragma unroll
    for (int s = 0; s < 4; ++s)
      loss += wgt[s] * (1.f - acc[s] * inv[s]);
    out[0] = loss;
  }
}

extern "C" void kernel_launch(void* const* d_in, const int* in_sizes, int n_in,
                              void* d_out, int out_size, void* d_ws, size_t ws_size,
                              hipStream_t stream)
{
  (void)in_sizes; (void)n_in; (void)out_size; (void)ws_size;

  const float* img1 = (const float*)d_in[0];
  const float* img2 = (const float*)d_in[1];
  float* out = (float*)d_out;
  float* ws  = (float*)d_ws;

  // workspace layout (floats): [0..3] scale accumulators, then pyramid levels
  float* acc = ws;
  float* l1a = ws + 16;
  float* l1b = l1a + 16 * 3 * 256 * 256;
  float* l2a = l1b + 16 * 3 * 256 * 256;
  float* l2b = l2a + 16 * 3 * 128 * 128;
  float* l3a = l2b + 16 * 3 * 128 * 128;
  float* l3b = l3a + 16 * 3 * 64 * 64;

  hipMemsetAsync(acc, 0, 4 * sizeof(float), stream);

  const int planes = 16 * 3;
  const dim3 blk(128);

  // scale 0: 512x512
  ms_ssim_scale_kernel<<<dim3(512 / TW, 512 / THT, planes), blk, 0, stream>>>(
      img1, img2, acc + 0, 512, 512);

  // scale 1: 256x256
  down2_kernel<<<dim3(256 / 64, 256 / 4, planes), dim3(256), 0, stream>>>(
      img1, img2, l1a, l1b, 256, 256);
  ms_ssim_scale_kernel<<<dim3(256 / TW, 256 / THT, planes), blk, 0, stream>>>(
      l1a, l1b, acc + 1, 256, 256);

  // scale 2: 128x128
  down2_kernel<<<dim3(128 / 64, 128 / 4, planes), dim3(256), 0, stream>>>(
      l1a, l1b, l2a, l2b, 128, 128);
  ms_ssim_scale_kernel<<<dim3(128 / TW, 128 / THT, planes), blk, 0, stream>>>(
      l2a, l2b, acc + 2, 128, 128);

  // scale 3: 64x64
  down2_kernel<<<dim3(64 / 64, 64 / 4, planes), dim3(256), 0, stream>>>(
      l2a, l2b, l3a, l3b, 64, 64);
  ms_ssim_scale_kernel<<<dim3(64 / TW, 64 / THT, planes), blk, 0, stream>>>(
      l3a, l3b, acc + 3, 64, 64);

  finalize_kernel<<<1, 32, 0, stream>>>(acc, out);
}